// MultiAspectRetrieval_23794118819888
// MI455X (gfx1250) — compile-verified
//
#include <hip/hip_runtime.h>
#include <math.h>

// Problem sizes (fixed by the reference)
#define B_SZ  512
#define N_SZ  8192
#define D_SZ  1024
#define DA_SZ 512
#define S_SZ  8
#define DK_SZ 64
#define SD_SZ 512   // S * D_K  (concatenated aspect dim)

typedef __attribute__((ext_vector_type(2))) float v2f;
typedef __attribute__((ext_vector_type(8))) float v8f;

// D = A(16x4 f32) x B(4x16 f32) + C(16x16 f32), full fp32 matrix pipe.
__device__ __forceinline__ v8f wmma_f32x4(v2f a, v2f b, v8f c) {
  return __builtin_amdgcn_wmma_f32_16x16x4_f32(
      /*neg_a=*/false, a, /*neg_b=*/false, b,
      /*c_mod=*/(short)0, c, /*reuse_a=*/false, /*reuse_b=*/false);
}

// ---------------------------------------------------------------------------
// Kernel 1: raw queries = z [512 x 512] @ concat(W_Q) [512 x 512] -> ws.qs
// W_Q layout [S, D_A, D_K]; column c maps to aspect c/64, in-aspect col c%64.
// Each wave: 4 M-tiles x 1 C-tile (B operand reused across M-tiles).
// ---------------------------------------------------------------------------
__global__ __launch_bounds__(256) void maq_gemm_q(const float* __restrict__ z,
                                                  const float* __restrict__ WQ,
                                                  float* __restrict__ qs) {
  const int wave = (blockIdx.x * 256 + threadIdx.x) >> 5;
  const int lane = threadIdx.x & 31;
  const int row  = lane & 15;
  const int half = lane >> 4;
  const int msuper = wave >> 5;   // 0..7   (64 rows each)
  const int ctile  = wave & 31;   // 0..31  (16 cols each)
  const int m0 = msuper * 64;
  const int c0 = ctile * 16;
  const float* Bp = WQ + (c0 >> 6) * (DA_SZ * DK_SZ) + (c0 & 63) + row;

  v8f acc[4] = {};
  for (int k0 = 0; k0 < DA_SZ; k0 += 4) {
    const int kk = k0 + 2 * half;
    v2f b;
    b[0] = Bp[(kk + 0) * DK_SZ];
    b[1] = Bp[(kk + 1) * DK_SZ];
#pragma unroll
    for (int t = 0; t < 4; ++t) {
      const float* ap = z + (m0 + t * 16 + row) * DA_SZ + kk;
      v2f a; a[0] = ap[0]; a[1] = ap[1];
      acc[t] = wmma_f32x4(a, b, acc[t]);
    }
  }
#pragma unroll
  for (int t = 0; t < 4; ++t)
#pragma unroll
    for (int i = 0; i < 8; ++i)
      qs[(m0 + t * 16 + i + 8 * half) * SD_SZ + c0 + row] = acc[t][i];
}

// ---------------------------------------------------------------------------
// Kernel 2: normalize each 64-wide aspect block of qs, fold in softmax weight.
// One thread per (b, aspect) row; float4 traffic.
// ---------------------------------------------------------------------------
__global__ __launch_bounds__(256) void maq_norm_q(float* __restrict__ qs,
                                                  const float* __restrict__ logits) {
  const int idx = blockIdx.x * 256 + threadIdx.x;   // < 4096
  const int b = idx >> 3;
  const int a = idx & 7;
  float mx = logits[0];
#pragma unroll
  for (int i = 1; i < S_SZ; ++i) mx = fmaxf(mx, logits[i]);
  float den = 0.0f;
#pragma unroll
  for (int i = 0; i < S_SZ; ++i) den += expf(logits[i] - mx);
  const float w = expf(logits[a] - mx) / den;

  float4* p = (float4*)(qs + b * SD_SZ + a * DK_SZ);
  float ss = 0.0f;
#pragma unroll
  for (int i = 0; i < DK_SZ / 4; ++i) {
    const float4 v = p[i];
    ss += v.x * v.x + v.y * v.y + v.z * v.z + v.w * v.w;
  }
  const float sc = w / (sqrtf(ss) + 1e-8f);
#pragma unroll
  for (int i = 0; i < DK_SZ / 4; ++i) {
    float4 v = p[i];
    v.x *= sc; v.y *= sc; v.z *= sc; v.w *= sc;
    p[i] = v;
  }
}

// ---------------------------------------------------------------------------
// Kernel 3: keys = pool [8192 x 1024] @ concat(W_K) [1024 x 512] -> d_out.keys
// Each wave: 4 M-tiles x 2 C-tiles (A loads shared across the C-tile pair)
// => 8 WMMA per (4 x b64 + 4 x b32) loads.
// ---------------------------------------------------------------------------
__global__ __launch_bounds__(256) void maq_gemm_k(const float* __restrict__ pool,
                                                  const float* __restrict__ WK,
                                                  float* __restrict__ keys) {
  const int wave = (blockIdx.x * 256 + threadIdx.x) >> 5;   // 0..2047
  const int lane = threadIdx.x & 31;
  const int row  = lane & 15;
  const int half = lane >> 4;
  const int msuper = wave >> 4;   // 0..127 (64 rows each)
  const int cpair  = wave & 15;   // 0..15  (32 cols each; both tiles same aspect)
  const int m0 = msuper * 64;
  const int c0 = cpair * 32;
  const float* Bp = WK + (c0 >> 6) * (D_SZ * DK_SZ) + (c0 & 63) + row;

  v8f acc[2][4] = {};
  for (int k0 = 0; k0 < D_SZ; k0 += 4) {
    const int kk = k0 + 2 * half;
    v2f b[2];
#pragma unroll
    for (int u = 0; u < 2; ++u) {
      b[u][0] = Bp[(kk + 0) * DK_SZ + u * 16];
      b[u][1] = Bp[(kk + 1) * DK_SZ + u * 16];
    }
#pragma unroll
    for (int t = 0; t < 4; ++t) {
      const float* ap = pool + (size_t)(m0 + t * 16 + row) * D_SZ + kk;
      v2f a; a[0] = ap[0]; a[1] = ap[1];
      acc[0][t] = wmma_f32x4(a, b[0], acc[0][t]);
      acc[1][t] = wmma_f32x4(a, b[1], acc[1][t]);
    }
  }
#pragma unroll
  for (int u = 0; u < 2; ++u)
#pragma unroll
    for (int t = 0; t < 4; ++t)
#pragma unroll
      for (int i = 0; i < 8; ++i)
        keys[(size_t)(m0 + t * 16 + i + 8 * half) * SD_SZ + c0 + u * 16 + row] =
            acc[u][t][i];
}

// ---------------------------------------------------------------------------
// Kernel 4: k_norm -> ws.kn. One thread per (n, aspect) row; float4 traffic.
// ---------------------------------------------------------------------------
__global__ __launch_bounds__(256) void maq_norm_k(const float* __restrict__ keys,
                                                  float* __restrict__ kn) {
  const int idx = blockIdx.x * 256 + threadIdx.x;   // < 65536
  const int n = idx >> 3;
  const int a = idx & 7;
  const float4* p = (const float4*)(keys + (size_t)n * SD_SZ + a * DK_SZ);
  float ss = 0.0f;
#pragma unroll
  for (int i = 0; i < DK_SZ / 4; ++i) {
    const float4 v = p[i];
    ss += v.x * v.x + v.y * v.y + v.z * v.z + v.w * v.w;
  }
  const float sc = 1.0f / (sqrtf(ss) + 1e-8f);
  float4* q = (float4*)(kn + (size_t)n * SD_SZ + a * DK_SZ);
#pragma unroll
  for (int i = 0; i < DK_SZ / 4; ++i) {
    float4 v = p[i];
    v.x *= sc; v.y *= sc; v.z *= sc; v.w *= sc;
    q[i] = v;
  }
}

// ---------------------------------------------------------------------------
// Kernel 5: scores = Q'[512 x 512] @ K'^T  (K' is [N x 512] row-major, so the
// WMMA B operand b[j] = kn[n][kk+j] is a contiguous float2 per lane).
// Each wave: 2 M-tiles x 4 N-tiles (B loads shared across the M-tile pair)
// => 8 WMMA per 6 x b64 loads.
// Fused epilogue: write scores, and alpha_raw = sigmoid(lam*(s-tau))*exp(s/T)
// into the alpha output section (normalized by kernels 6+7).
// ---------------------------------------------------------------------------
__global__ __launch_bounds__(256) void maq_gemm_scores(
    const float* __restrict__ qs, const float* __restrict__ kn,
    const float* __restrict__ tauP, const float* __restrict__ lamP,
    const float* __restrict__ tmpP,
    float* __restrict__ scores, float* __restrict__ alphaRaw) {
  const int wave = (blockIdx.x * 256 + threadIdx.x) >> 5;   // 0..2047
  const int lane = threadIdx.x & 31;
  const int row  = lane & 15;
  const int half = lane >> 4;
  const int mpair  = wave >> 7;    // 0..15  (32 rows each)
  const int nsuper = wave & 127;   // 0..127 (64 cols each)
  const int m0 = mpair * 32;
  const int n0 = nsuper * 64;

  v8f acc[2][4] = {};
  for (int k0 = 0; k0 < SD_SZ; k0 += 4) {
    const int kk = k0 + 2 * half;
    v2f a[2];
#pragma unroll
    for (int u = 0; u < 2; ++u) {
      const float* ap = qs + (m0 + u * 16 + row) * SD_SZ + kk;
      a[u][0] = ap[0]; a[u][1] = ap[1];
    }
#pragma unroll
    for (int t = 0; t < 4; ++t) {
      const float* kp = kn + (size_t)(n0 + t * 16 + row) * SD_SZ + kk;
      v2f b; b[0] = kp[0]; b[1] = kp[1];
      acc[0][t] = wmma_f32x4(a[0], b, acc[0][t]);
      acc[1][t] = wmma_f32x4(a[1], b, acc[1][t]);
    }
  }

  const float tau  = *tauP;
  const float lam  = *lamP;
  const float invT = 1.0f / *tmpP;
#pragma unroll
  for (int u = 0; u < 2; ++u) {
#pragma unroll
    for (int t = 0; t < 4; ++t) {
#pragma unroll
      for (int i = 0; i < 8; ++i) {
        const int m = m0 + u * 16 + i + 8 * half;
        const int n = n0 + t * 16 + row;
        const float s = acc[u][t][i];
        scores[(size_t)m * N_SZ + n] = s;
        const float g = 1.0f / (1.0f + expf(-lam * (s - tau)));
        alphaRaw[(size_t)m * N_SZ + n] = g * expf(s * invT);
      }
    }
  }
}

// ---------------------------------------------------------------------------
// Kernel 6: deterministic per-row sum of alpha_raw (one block per row).
// ---------------------------------------------------------------------------
__global__ __launch_bounds__(256) void maq_rowsum(const float* __restrict__ ar,
                                                  float* __restrict__ sums) {
  __shared__ float red[256];
  const int b = blockIdx.x;
  const float4* p = (const float4*)(ar + (size_t)b * N_SZ);
  float s = 0.0f;
  for (int i = threadIdx.x; i < N_SZ / 4; i += 256) {
    const float4 v = p[i];
    s += v.x + v.y + v.z + v.w;
  }
  red[threadIdx.x] = s;
  __syncthreads();
  for (int off = 128; off > 0; off >>= 1) {
    if (threadIdx.x < off) red[threadIdx.x] += red[threadIdx.x + off];
    __syncthreads();
  }
  if (threadIdx.x == 0) sums[b] = red[0];
}

// ---------------------------------------------------------------------------
// Kernel 7: alpha = alpha_raw / (rowsum + 1e-8), float4 per thread.
// ---------------------------------------------------------------------------
__global__ __launch_bounds__(256) void maq_scale_alpha(float4* __restrict__ alpha,
                                                       const float* __restrict__ sums) {
  const size_t idx = (size_t)blockIdx.x * 256 + threadIdx.x;  // over N*B/4 float4s
  const float inv = 1.0f / (sums[idx >> 11] + 1e-8f);         // 2048 float4 per row
  float4 v = alpha[idx];
  v.x *= inv; v.y *= inv; v.z *= inv; v.w *= inv;
  alpha[idx] = v;
}

// ---------------------------------------------------------------------------
extern "C" void kernel_launch(void* const* d_in, const int* in_sizes, int n_in,
                              void* d_out, int out_size, void* d_ws, size_t ws_size,
                              hipStream_t stream) {
  const float* z      = (const float*)d_in[0];  // [512, 512]
  const float* pool   = (const float*)d_in[1];  // [8192, 1024]
  const float* WQ     = (const float*)d_in[2];  // [8, 512, 64]
  const float* WK     = (const float*)d_in[3];  // [8, 1024, 64]
  const float* logits = (const float*)d_in[4];  // [8]
  const float* tau    = (const float*)d_in[5];
  const float* lam    = (const float*)d_in[6];
  const float* temp   = (const float*)d_in[7];

  float* out    = (float*)d_out;
  float* alpha  = out;                                  // [512, 8192]
  float* scores = out + (size_t)B_SZ * N_SZ;            // [512, 8192]
  float* keys   = out + (size_t)2 * B_SZ * N_SZ;        // [8192, 8, 64]

  float* ws = (float*)d_ws;
  float* qs = ws;                                       // [512, 512]   1 MB
  float* kn = qs + (size_t)B_SZ * SD_SZ;                // [8192, 512] 16 MB
  float* rs = kn + (size_t)N_SZ * SD_SZ;                // [512]

  // 1) raw queries (256 waves)
  maq_gemm_q<<<32, 256, 0, stream>>>(z, WQ, qs);
  // 2) normalize + aspect-softmax weight fold (4096 threads)
  maq_norm_q<<<16, 256, 0, stream>>>(qs, logits);
  // 3) keys projection straight into d_out (2048 waves, 4x2 tiles each)
  maq_gemm_k<<<256, 256, 0, stream>>>(pool, WK, keys);
  // 4) normalized keys into workspace (65536 threads)
  maq_norm_k<<<256, 256, 0, stream>>>(keys, kn);
  // 5) scores GEMM + fused gate/exp epilogue (2048 waves, 2x4 tiles each)
  maq_gemm_scores<<<256, 256, 0, stream>>>(qs, kn, tau, lam, temp, scores, alpha);
  // 6) deterministic row sums
  maq_rowsum<<<512, 256, 0, stream>>>(alpha, rs);
  // 7) normalize alpha (1,048,576 float4s)
  maq_scale_alpha<<<4096, 256, 0, stream>>>((float4*)alpha, rs);
}